// TVLoss_7086696038732
// MI455X (gfx1250) — compile-verified
//
#include <hip/hip_runtime.h>

typedef __attribute__((ext_vector_type(2))) float v2f;
typedef __attribute__((ext_vector_type(8))) float v8f;
typedef __attribute__((ext_vector_type(4))) float f4v;

#define TV_H 256
#define TV_W 256
#define TV_BLOCKS 2048
#define TV_THREADS 256

// Sum all values held in a wave's v2f A-fragment using the fp32 WMMA:
// D = A(16x4) * ones(4x16); rowsum_m replicated across each row of D.
// Lanes 0..15 see rows 0..7 in d[0..7], lanes 16..31 see rows 8..15, so
// (sum of d[0..7] at lane 0) + (same at lane 16) == sum of all 64 A entries.
__device__ __forceinline__ v8f wmma_rowsum(v2f a) {
    v2f b; b.x = 1.0f; b.y = 1.0f;
    v8f c = {};
    return __builtin_amdgcn_wmma_f32_16x16x4_f32(
        /*neg_a=*/false, a, /*neg_b=*/false, b,
        /*c_mod=*/(short)0, c, /*reuse_a=*/false, /*reuse_b=*/false);
}

// Pass 1: stencil + per-block partial sums. Grid-stride over groups of 4 rows.
__global__ void tv_circgrad_partial(const float* __restrict__ f,
                                    float* __restrict__ partials,
                                    int groups) {
    const int tid  = threadIdx.x;
    const int lane = tid & 31;
    const int wave = tid >> 5;
    const int lr   = tid >> 6;          // row within 4-row group (0..3)
    const int xb   = (tid & 63) << 2;   // x base, 4 elements per thread

    float acc = 0.0f;

    for (int g = blockIdx.x; g < groups; g += TV_BLOCKS) {
        const int p = g >> 6;                       // 64 groups per 256-row plane
        const int y = ((g & 63) << 2) + lr;
        const float* plane = f + (size_t)p * (TV_H * TV_W);
        const int yb = (y + 1) & (TV_H - 1);

        f4v v  = *(const f4v*)(plane + y  * TV_W + xb);
        f4v bv = *(const f4v*)(plane + yb * TV_W + xb);
        float xn = plane[y * TV_W + ((xb + 4) & (TV_W - 1))];

        // Prefetch next grid-stride tile (emits global_prefetch_b8).
        int g2 = g + TV_BLOCKS;
        if (g2 < groups) {
            int p2 = g2 >> 6;
            int y2 = ((g2 & 63) << 2) + lr;
            __builtin_prefetch(f + (size_t)p2 * (TV_H * TV_W) + y2 * TV_W + xb, 0, 1);
        }

        float gx0 = v.x - v.y, gx1 = v.y - v.z, gx2 = v.z - v.w, gx3 = v.w - xn;
        float gy0 = v.x - bv.x, gy1 = v.y - bv.y, gy2 = v.z - bv.z, gy3 = v.w - bv.w;

        acc += sqrtf(gx0 * gx0 + gy0 * gy0);
        acc += sqrtf(gx1 * gx1 + gy1 * gy1);
        acc += sqrtf(gx2 * gx2 + gy2 * gy2);
        acc += sqrtf(gx3 * gx3 + gy3 * gy3);
    }

    // Per-wave reduction via fp32 WMMA (EXEC is full here: loop reconverged).
    v2f a; a.x = acc; a.y = 0.0f;
    v8f d = wmma_rowsum(a);
    float s = 0.0f;
#pragma unroll
    for (int i = 0; i < 8; ++i) s += d[i];

    __shared__ float lsum[16];          // 8 waves * 2 halves
    if ((lane & 15) == 0) lsum[wave * 2 + (lane >> 4)] = s;
    __syncthreads();

    if (tid == 0) {
        float t = 0.0f;
#pragma unroll
        for (int i = 0; i < 16; ++i) t += lsum[i];
        partials[blockIdx.x] = t;       // deterministic: fixed order, no atomics
    }
}

// Pass 2: one wave reduces TV_BLOCKS partials; final WMMA fp32 tree step.
__global__ void tv_circgrad_final(const float* __restrict__ partials,
                                  float* __restrict__ out,
                                  float inv_n) {
    const int lane = threadIdx.x;       // 32 threads

    v2f a; a.x = 0.0f; a.y = 0.0f;
    // TV_BLOCKS (2048) divisible by 64 -> uniform trip count, EXEC stays full.
    for (int i = lane; i < TV_BLOCKS; i += 64)       a.x += partials[i];
    for (int i = lane + 32; i < TV_BLOCKS; i += 64)  a.y += partials[i];

    v8f d = wmma_rowsum(a);
    float s = 0.0f;
#pragma unroll
    for (int i = 0; i < 8; ++i) s += d[i];

    __shared__ float halves[2];
    if ((lane & 15) == 0) halves[lane >> 4] = s;
    __syncthreads();
    if (lane == 0) out[0] = (halves[0] + halves[1]) * inv_n;
}

extern "C" void kernel_launch(void* const* d_in, const int* in_sizes, int n_in,
                              void* d_out, int out_size, void* d_ws, size_t ws_size,
                              hipStream_t stream) {
    const float* f = (const float*)d_in[0];
    float* out = (float*)d_out;
    float* partials = (float*)d_ws;     // needs TV_BLOCKS * 4 bytes = 8 KB

    const int n = in_sizes[0];                 // B*C*T*H*W = 25,165,824
    const int planes = n / (TV_H * TV_W);      // 384
    const int groups = planes * (TV_H / 4);    // 24,576 four-row groups
    const float inv_n = 1.0f / (float)n;

    tv_circgrad_partial<<<TV_BLOCKS, TV_THREADS, 0, stream>>>(f, partials, groups);
    tv_circgrad_final<<<1, 32, 0, stream>>>(partials, out, inv_n);
}